// GCN_Layer_55860344652275
// MI455X (gfx1250) — compile-verified
//
#include <hip/hip_runtime.h>

#define D_FEAT 128
#define BATCH  32          // edges staged per wave per buffer
#define GRP    8           // edges processed per manually-pipelined group
#define WPB    8           // waves per block (wave32)
#define BLOCK  (WPB * 32)

// Low 32 bits of a generic pointer to __shared__ = workgroup-relative LDS
// byte offset (flat LDS aperture).
__device__ __forceinline__ unsigned lds_off(const void* p) {
  return (unsigned)(unsigned long long)p;
}

// CDNA5 async copy: global dword at vaddr copied into LDS at vdst; tracked with ASYNCcnt.
// Edge metadata is streamed exactly once -> non-temporal hint so it does not
// evict the L2-resident feature matrix (51.2 MB) + output (51.2 MB).
__device__ __forceinline__ void async_b32_nt(unsigned ldsoff, const void* gaddr) {
  unsigned long long a = (unsigned long long)gaddr;
  asm volatile("global_load_async_to_lds_b32 %0, %1, off th:TH_LOAD_NT"
               :: "v"(ldsoff), "v"(a) : "memory");
}

// Hardware no-return f32 atomic adds (STOREcnt; s_endpgm implicitly waits idle).
// NO memory clobber: nothing in this kernel reads 'out'; subsequent gather
// loads may (and should) issue underneath these fire-and-forget atomics.
__device__ __forceinline__ void atomic_add_row16(float* base, float4 x) {
  unsigned long long a = (unsigned long long)base;
  asm volatile("global_atomic_add_f32 %0, %1, off scope:SCOPE_DEV"
               :: "v"(a), "v"(x.x));
  asm volatile("global_atomic_add_f32 %0, %1, off offset:4 scope:SCOPE_DEV"
               :: "v"(a), "v"(x.y));
  asm volatile("global_atomic_add_f32 %0, %1, off offset:8 scope:SCOPE_DEV"
               :: "v"(a), "v"(x.z));
  asm volatile("global_atomic_add_f32 %0, %1, off offset:12 scope:SCOPE_DEV"
               :: "v"(a), "v"(x.w));
}

// ---------------------------------------------------------------------------
// SpMM scatter: one wave32 per edge; 32 lanes x float4 = one 128-f32 row.
// Edge metadata double-buffered in LDS via async-to-LDS loads (ASYNCcnt).
// ---------------------------------------------------------------------------
__global__ __launch_bounds__(BLOCK) void gcn_spmm(
    const int*   __restrict__ ei,    // [2][E]: row0 = dst, row1 = src
    const float* __restrict__ ev,    // [E]
    const float* __restrict__ feat,  // [N][128]
    float*       __restrict__ out,   // [N][128]
    int E) {
  // per wave: 2 buffers x (dst[32] | src[32] | valbits[32]); 16B aligned so
  // adjacent uniform metadata reads can merge into ds_load_b64/b128.
  __shared__ __align__(16) int smem[WPB * 2 * 3 * BATCH];

  const int lane = threadIdx.x & 31;
  const int wave = threadIdx.x >> 5;
  const long long nBatches = ((long long)E + BATCH - 1) / BATCH;
  const long long waveId   = (long long)blockIdx.x * WPB + wave;
  const long long nWaves   = (long long)gridDim.x * WPB;

  int* wbase = smem + wave * (2 * 3 * BATCH);

  auto stage = [&](int buf, long long b) {
    long long e = b * BATCH + lane;
    if (e < E) {
      int* s = wbase + buf * (3 * BATCH);
      async_b32_nt(lds_off(s + lane),             ei + e);      // dst
      async_b32_nt(lds_off(s + BATCH + lane),     ei + E + e);  // src
      async_b32_nt(lds_off(s + 2 * BATCH + lane), ev + e);      // val bits
    }
  };

  const float4* __restrict__ feat4 = (const float4*)feat;

  if (waveId >= nBatches) return;

  int buf = 0;
  stage(0, waveId);                      // prologue: batch 0 in flight (3 asyncs)
  for (long long b = waveId; b < nBatches; b += nWaves, buf ^= 1) {
    long long nb = b + nWaves;
    if (nb < nBatches) {
      stage(buf ^ 1, nb);                               // next batch in flight
      asm volatile("s_wait_asynccnt 3" ::: "memory");   // drain current 3 only
    } else {
      asm volatile("s_wait_asynccnt 0" ::: "memory");
    }

    const int* s = wbase + buf * (3 * BATCH);
    long long rem = E - b * (long long)BATCH;

    if (rem >= BATCH) {
      // Hot path: full batch, compile-time structure, manual 8-wide pipeline.
#pragma unroll
      for (int k = 0; k < BATCH; k += GRP) {
        if (k + GRP < BATCH) {   // warm L2->WGP path for the next group
          __builtin_prefetch(
              feat + (size_t)s[BATCH + k + GRP] * D_FEAT + lane * 4, 0, 0);
        }
        int   d[GRP], sr[GRP];
        float vv[GRP];
#pragma unroll
        for (int j = 0; j < GRP; ++j) {
          d[j]  = s[k + j];                       // uniform-address LDS reads
          sr[j] = s[BATCH + k + j];
          vv[j] = __int_as_float(s[2 * BATCH + k + j]);
        }
        float4 f[GRP];
#pragma unroll
        for (int j = 0; j < GRP; ++j)             // 8 independent b128 gathers
          f[j] = feat4[(size_t)sr[j] * (D_FEAT / 4) + lane];
#pragma unroll
        for (int j = 0; j < GRP; ++j) {           // 32 fire-and-forget atomics
          float4 m = make_float4(vv[j] * f[j].x, vv[j] * f[j].y,
                                 vv[j] * f[j].z, vv[j] * f[j].w);
          atomic_add_row16(out + (size_t)d[j] * D_FEAT + lane * 4, m);
        }
      }
    } else {
      // Tail batch (never taken when E % 32 == 0): simple per-edge loop.
      int cnt = (int)rem;
      for (int k = 0; k < cnt; ++k) {
        int   dst = s[k];
        int   src = s[BATCH + k];
        float v   = __int_as_float(s[2 * BATCH + k]);
        float4 f  = feat4[(size_t)src * (D_FEAT / 4) + lane];
        float4 m  = make_float4(v * f.x, v * f.y, v * f.z, v * f.w);
        atomic_add_row16(out + (size_t)dst * D_FEAT + lane * 4, m);
      }
    }
  }
}

// ---------------------------------------------------------------------------
// Zero-fill output (harness poisons d_out with 0xAA; we accumulate atomically)
// ---------------------------------------------------------------------------
__global__ __launch_bounds__(256) void gcn_zero(float4* __restrict__ out, int n4) {
  int i = blockIdx.x * blockDim.x + threadIdx.x;
  int stride = gridDim.x * blockDim.x;
  for (; i < n4; i += stride) out[i] = make_float4(0.f, 0.f, 0.f, 0.f);
}

// ---------------------------------------------------------------------------
extern "C" void kernel_launch(void* const* d_in, const int* in_sizes, int n_in,
                              void* d_out, int out_size, void* d_ws, size_t ws_size,
                              hipStream_t stream) {
  const int*   ei   = (const int*)d_in[0];    // edge_index [2][E]
  const float* ev   = (const float*)d_in[1];  // edge_vals [E]
  const float* feat = (const float*)d_in[2];  // features [N][128]
  float*       out  = (float*)d_out;

  const int E  = in_sizes[1];
  const int n4 = out_size / 4;

  gcn_zero<<<1024, 256, 0, stream>>>((float4*)out, n4);

  long long nBatches = ((long long)E + BATCH - 1) / BATCH;
  long long wantBlocks = (nBatches + WPB - 1) / WPB;
  int blocks = (int)(wantBlocks < 1250 ? wantBlocks : 1250);  // ~10 batches/wave
  if (blocks < 1) blocks = 1;

  gcn_spmm<<<blocks, BLOCK, 0, stream>>>(ei, ev, feat, out, E);
}